// TsdfRFRenderer_79706003079352
// MI455X (gfx1250) — compile-verified
//
#include <hip/hip_runtime.h>
#include <cstdint>
#include <cstddef>

#define NRAYS  65536
#define SU     256      // coarse samples per ray
#define NIMP   128      // importance samples per ray
#define RPB    16       // rays per block
#define TPB    256      // 8 waves of 32
#define OUTW   132      // 3 rgb + 1 depth + 128 weights
#define TRUNCV 0.015625f  // 2/512*4

typedef float v2f __attribute__((ext_vector_type(2)));
typedef float v8f __attribute__((ext_vector_type(8)));

#if defined(__AMDGCN__) && __has_builtin(__builtin_amdgcn_wmma_f32_16x16x4_f32)
#define HAS_WMMA4 1
#endif

#if defined(__AMDGCN__) && __has_builtin(__builtin_amdgcn_global_load_async_to_lds_b128)
#define HAS_ASYNC_LDS 1
#endif

#ifdef HAS_ASYNC_LDS
// Builtin prototype (from compiler diagnostic):
//   void __builtin_amdgcn_global_load_async_to_lds_b128(
//       int __attribute__((vector_size(16))) addrspace(1)* gaddr,
//       int __attribute__((vector_size(16))) addrspace(3)* lds,
//       imm int offset, imm int cpol);
typedef int v4i_raw __attribute__((__vector_size__(16)));
typedef __attribute__((address_space(1))) v4i_raw gas_v4i;
typedef __attribute__((address_space(3))) v4i_raw las_v4i;

__device__ __forceinline__ gas_v4i* as_global(const void* p) {
    // int->AS(1) pointer cast (value-preserving), avoids addrspacecast restrictions
    return (gas_v4i*)(uintptr_t)p;
}
__device__ __forceinline__ las_v4i* as_lds(const void* p) {
    // low 32 bits of a generic LDS pointer are the LDS byte offset
    return (las_v4i*)(uint32_t)(uintptr_t)p;
}
__device__ __forceinline__ void wait_async0() {
#if __has_builtin(__builtin_amdgcn_s_wait_asynccnt)
    __builtin_amdgcn_s_wait_asynccnt(0);
#else
    asm volatile("s_wait_asynccnt 0x0" ::: "memory");
#endif
}
#endif

__device__ __forceinline__ float sigmoidf_(float x) {
    return 1.0f / (1.0f + __expf(-x));
}

__device__ __forceinline__ float wave_fadd(float v) {
    #pragma unroll
    for (int m = 16; m >= 1; m >>= 1) v += __shfl_xor(v, m, 32);
    return v;
}

__device__ __forceinline__ int wave_imin(int v) {
    #pragma unroll
    for (int m = 16; m >= 1; m >>= 1) {
        int o = __shfl_xor(v, m, 32);
        v = (o < v) ? o : v;
    }
    return v;
}

__global__ __launch_bounds__(TPB)
void TsdfRFRenderer_kernel(const float* __restrict__ occ,
                           const float* __restrict__ zvals,
                           const float* __restrict__ sdf,
                           const float* __restrict__ rgbs,
                           float* __restrict__ out)
{
    // LDS layout (floats):
    __shared__ float sPdf[SU * RPB];   // [pos 0..255][ray]; pos 254/255 zero-padded.
                                       // Reused after phase 2a: [0..2047] z_samples[ray][i],
                                       //                        [2048..4095] staged sdf[ray][i]
    __shared__ float sCdf[SU * RPB];   // inclusive prefix sums, [pos][ray]
    __shared__ float sZ[RPB * SU];     // z_vals_u, [ray][s]
    __shared__ float sCp[RPB * 16];    // 16-wide chunk products per ray
    __shared__ float sCb[16 * RPB];    // per-chunk exclusive base, [chunk][ray]
    __shared__ float sTot[RPB];        // per-ray pdf total

    const int t    = threadIdx.x;
    const int r    = t >> 4;          // local ray 0..15 (phases 1,3)
    const int q    = t & 15;          // 16 threads per ray
    const int rayg = blockIdx.x * RPB + r;

    const int wv   = t >> 5;          // wave 0..7
    const int ln   = t & 31;
    const int col  = ln & 15;         // WMMA column (= ray)
    const int half = ln >> 4;

    float* sZs  = sPdf;                 // z_samples [ray][i], 16*128 (phase >= 3)
    float* sSdf = sPdf + RPB * NIMP;    // staged sdf [ray][i], 16*128 (phase >= 3)

    // Prefetch the rgbs lines this thread's wave consumes in phase 4.
    {
        const size_t rgbase = ((size_t)blockIdx.x * RPB + wv * 2) * NIMP * 3 + (size_t)(ln * 4) * 3;
        __builtin_prefetch(rgbs + rgbase, 0, 1);
        __builtin_prefetch(rgbs + rgbase + (size_t)NIMP * 3, 0, 1);
    }

    // ================= Phase 1: occupancy -> pdf (position-major in LDS) ==========
    // z_vals: pure global->LDS move, no ALU -> async copy engine path.
    const float* gz = zvals + (size_t)rayg * SU + q * 16;   // 64 B per thread
#ifdef HAS_ASYNC_LDS
    {
        gas_v4i* g = as_global(gz);
        las_v4i* l = as_lds(&sZ[r * SU + q * 16]);
        __builtin_amdgcn_global_load_async_to_lds_b128(g, l, 0,  0);
        __builtin_amdgcn_global_load_async_to_lds_b128(g, l, 16, 0);
        __builtin_amdgcn_global_load_async_to_lds_b128(g, l, 32, 0);
        __builtin_amdgcn_global_load_async_to_lds_b128(g, l, 48, 0);
    }
#endif

    float av[16];   // alpha per element of this thread's 16-chunk
    float pl[16];   // exclusive in-chunk prefix product of (1-alpha+1e-10)
    {
        const float4* po = reinterpret_cast<const float4*>(occ + (size_t)rayg * SU + q * 16);
#ifndef HAS_ASYNC_LDS
        const float4* pz = reinterpret_cast<const float4*>(gz);
#endif
        float prod = 1.0f;
        #pragma unroll
        for (int v4 = 0; v4 < 4; ++v4) {
            float4 o4 = po[v4];
            float ov[4] = {o4.x, o4.y, o4.z, o4.w};
#ifndef HAS_ASYNC_LDS
            float4 z4 = pz[v4];
            float zv[4] = {z4.x, z4.y, z4.z, z4.w};
#endif
            #pragma unroll
            for (int k = 0; k < 4; ++k) {
                int kk = v4 * 4 + k;
                float a = sigmoidf_(10.0f * ov[k]);
                av[kk] = a;
                pl[kk] = prod;
                prod *= (1.0f - a + 1e-10f);
#ifndef HAS_ASYNC_LDS
                sZ[r * SU + q * 16 + kk] = zv[k];
#endif
            }
        }
        sCp[r * 16 + q] = prod;
    }
    __syncthreads();
    {
        float base = 1.0f;
        for (int j = 0; j < q; ++j) base *= sCp[r * 16 + j];
        #pragma unroll
        for (int kk = 0; kk < 16; ++kk) {
            int s = q * 16 + kk;
            float trans = base * pl[kk];
            float wu = av[kk] * trans;
            if (s >= 1 && s <= 254) sPdf[(s - 1) * RPB + r] = wu + 1e-5f;  // pdf index s-1
        }
        if (q == 15) {  // pad positions 254,255 (only 254 real pdf entries)
            sPdf[254 * RPB + r] = 0.0f;
            sPdf[255 * RPB + r] = 0.0f;
        }
    }
    __syncthreads();

    // ================= Phase 2a: intra-chunk inclusive prefix via WMMA ============
    // D = L * X : L = 16x16 lower-triangular ones (A, split into 4 K-slices of 16x4),
    // X = 16-position pdf chunk for 16 rays (B, 4x16 per slice), D[m][ray] = prefix.
    for (int c = wv * 2; c < wv * 2 + 2; ++c) {
#ifdef HAS_WMMA4
        v8f acc = {0.f, 0.f, 0.f, 0.f, 0.f, 0.f, 0.f, 0.f};
        #pragma unroll
        for (int kc = 0; kc < 4; ++kc) {
            const int K0 = kc * 4 + half * 2;   // A/B K indices held by this lane
            const int m  = col;                 // A row = lane % 16
            v2f a, b;
            a.x = (K0     <= m) ? 1.0f : 0.0f;
            a.y = (K0 + 1 <= m) ? 1.0f : 0.0f;
            b.x = sPdf[(c * 16 + K0    ) * RPB + col];
            b.y = sPdf[(c * 16 + K0 + 1) * RPB + col];
            acc = __builtin_amdgcn_wmma_f32_16x16x4_f32(
                      false, a, false, b, (short)0, acc, false, false);
        }
        #pragma unroll
        for (int j = 0; j < 8; ++j) {
            int m = j + half * 8;               // C/D: lanes<16 rows 0..7, lanes>=16 rows 8..15
            sCdf[(c * 16 + m) * RPB + col] = acc[j];
        }
#else
        // VALU fallback: same math, serial per row
        for (int j = 0; j < 8; ++j) {
            int m = j + half * 8;
            float s = 0.0f;
            for (int k = 0; k <= m; ++k) s += sPdf[(c * 16 + k) * RPB + col];
            sCdf[(c * 16 + m) * RPB + col] = s;
        }
#endif
    }
    __syncthreads();

#ifdef HAS_ASYNC_LDS
    // sdf staging: pure global->LDS move into the (now dead) upper half of sPdf.
    // Must be issued only after ALL waves finished reading sPdf in phase 2a.
    {
        gas_v4i* g = as_global(sdf + (size_t)rayg * NIMP + q * 8);   // 32 B per thread
        las_v4i* l = as_lds(&sSdf[r * NIMP + q * 8]);
        __builtin_amdgcn_global_load_async_to_lds_b128(g, l, 0,  0);
        __builtin_amdgcn_global_load_async_to_lds_b128(g, l, 16, 0);
    }
#endif

    // ================= Phase 2b: chunk-base scan per ray ==========================
    if (t < RPB) {
        float off = 0.0f;
        for (int c = 0; c < 16; ++c) {
            sCb[c * RPB + t] = off;
            off += sCdf[(c * 16 + 15) * RPB + t];
        }
        sTot[t] = off;
    }
    __syncthreads();

    // ================= Phase 2c: add bases -> global inclusive sums ===============
    for (int c = wv * 2; c < wv * 2 + 2; ++c) {
        float base = sCb[c * RPB + col];
        #pragma unroll
        for (int j = 0; j < 8; ++j) {
            int m = j + half * 8;
            sCdf[(c * 16 + m) * RPB + col] += base;
        }
    }
#ifdef HAS_ASYNC_LDS
    wait_async0();   // z (phase 1) + sdf (post-2a) async copies resident in LDS
#endif
    __syncthreads();

    // ================= Phase 3: inverse-CDF sampling (+ sdf staging fallback) =====
    {
#ifndef HAS_ASYNC_LDS
        const float4* ps = reinterpret_cast<const float4*>(sdf + (size_t)rayg * NIMP + q * 8);
        float4 s0 = ps[0], s1 = ps[1];
        float sv[8] = {s0.x, s0.y, s0.z, s0.w, s1.x, s1.y, s1.z, s1.w};
        #pragma unroll
        for (int k = 0; k < 8; ++k) sSdf[r * NIMP + q * 8 + k] = sv[k];
#endif
        const float tot  = sTot[r];
        const float tinv = 1.0f / tot;
        #pragma unroll 2
        for (int ii = 0; ii < 8; ++ii) {
            int   i  = q * 8 + ii;
            float u  = (float)i * (1.0f / 127.0f);
            float ut = u * tot;   // compare unnormalized: S[k] <= u*total  <=>  cdf <= u
            // count k in [0,253] with S[k] <= ut  (cdf entries 1..254; cdf[0]=0 always <= u)
            int lo = 0, hi = 254;
            while (lo < hi) {
                int mid = (lo + hi) >> 1;
                if (sCdf[mid * RPB + r] <= ut) lo = mid + 1; else hi = mid;
            }
            const int below = lo;                              // inds-1, in [0,254]
            const int above = (lo + 1 < 254) ? (lo + 1) : 254; // min(inds,254)
            float Sb = (below == 0) ? 0.0f : sCdf[(below - 1) * RPB + r];
            float Sa = (above == 0) ? 0.0f : sCdf[(above - 1) * RPB + r];
            float cb = Sb * tinv, ca = Sa * tinv;
            float den = ca - cb;
            if (den < 1e-5f) den = 1.0f;                       // reference denom guard
            float tt = (u - cb) / den;
            float binb = 0.5f * (sZ[r * SU + below] + sZ[r * SU + below + 1]);
            float bina = 0.5f * (sZ[r * SU + above] + sZ[r * SU + above + 1]);
            sZs[r * NIMP + i] = binb + tt * (bina - binb);
        }
    }
    __syncthreads();

    // ================= Phase 4: TSDF weights + composite (1 wave per 2 rays) ======
    for (int rr = wv * 2; rr < wv * 2 + 2; ++rr) {
        const int rg = blockIdx.x * RPB + rr;
        const int i0 = ln * 4;

        float sd[4], zz[4];
        #pragma unroll
        for (int k = 0; k < 4; ++k) {
            sd[k] = sSdf[rr * NIMP + i0 + k];
            zz[k] = sZs [rr * NIMP + i0 + k];
        }
        float sd4 = (i0 + 4 < NIMP) ? sSdf[rr * NIMP + i0 + 4] : 0.0f;

        // first zero-crossing of sdf (argmax of sign mask == first j with product < 0)
        int fmin = 0x7fffffff;
        #pragma unroll
        for (int k = 0; k < 4; ++k) {
            int j = i0 + k;
            if (j < NIMP - 1) {
                float p = sd[k] * ((k < 3) ? sd[k + 1] : sd4);
                if (p < 0.0f && j < fmin) fmin = j;
            }
        }
        fmin = wave_imin(fmin);
        const int   firstidx = (fmin == 0x7fffffff) ? 0 : fmin;
        const float thr = sZs[rr * NIMP + firstidx] + TRUNCV;

        float al[4], asum = 0.0f;
        #pragma unroll
        for (int k = 0; k < 4; ++k) {
            float x = sd[k] * (1.0f / TRUNCV);
            float s = sigmoidf_(x);
            float a = s * (1.0f - s);           // sigmoid(x)*sigmoid(-x)
            if (!(zz[k] < thr)) a = 0.0f;       // truncation mask
            al[k] = a;
            asum += a;
        }
        asum = wave_fadd(asum) + 1e-8f;
        const float winv = 1.0f / asum;

        const float4* pr = reinterpret_cast<const float4*>(rgbs + (size_t)rg * NIMP * 3 + (size_t)i0 * 3);
        float4 c0 = pr[0], c1 = pr[1], c2 = pr[2];
        float rc[12] = {c0.x, c0.y, c0.z, c0.w, c1.x, c1.y, c1.z, c1.w, c2.x, c2.y, c2.z, c2.w};

        float R = 0.0f, G = 0.0f, B = 0.0f, D = 0.0f;
        #pragma unroll
        for (int k = 0; k < 4; ++k) {
            float w = al[k] * winv;
            D += w * zz[k];
            R += w * rc[3 * k + 0];
            G += w * rc[3 * k + 1];
            B += w * rc[3 * k + 2];
            out[(size_t)rg * OUTW + 4 + i0 + k] = w;
        }
        R = wave_fadd(R); G = wave_fadd(G); B = wave_fadd(B); D = wave_fadd(D);
        if (ln == 0) {
            out[(size_t)rg * OUTW + 0] = fminf(fmaxf(R, 0.0f), 1.0f);
            out[(size_t)rg * OUTW + 1] = fminf(fmaxf(G, 0.0f), 1.0f);
            out[(size_t)rg * OUTW + 2] = fminf(fmaxf(B, 0.0f), 1.0f);
            out[(size_t)rg * OUTW + 3] = D;
        }
    }
}

extern "C" void kernel_launch(void* const* d_in, const int* in_sizes, int n_in,
                              void* d_out, int out_size, void* d_ws, size_t ws_size,
                              hipStream_t stream) {
    (void)in_sizes; (void)n_in; (void)out_size; (void)d_ws; (void)ws_size;
    const float* occ  = (const float*)d_in[0];
    const float* zu   = (const float*)d_in[1];
    const float* sdf  = (const float*)d_in[2];
    const float* rgbs = (const float*)d_in[3];
    float* out = (float*)d_out;

    dim3 grid(NRAYS / RPB);   // 4096 blocks x 256 threads (8 wave32)
    TsdfRFRenderer_kernel<<<grid, TPB, 0, stream>>>(occ, zu, sdf, rgbs, out);
}